// GCN_42838003810368
// MI455X (gfx1250) — compile-verified
//
#include <hip/hip_runtime.h>
#include <math.h>

typedef __attribute__((ext_vector_type(2))) float v2f;
typedef __attribute__((ext_vector_type(8))) float v8f;

#define ATOMIC_ADD_F32(p, v) \
  __hip_atomic_fetch_add((p), (v), __ATOMIC_RELAXED, __HIP_MEMORY_SCOPE_AGENT)

// ---------------------------------------------------------------------------
// deg[i] = 1.0 (self loop)
__global__ __launch_bounds__(256) void deg_init_k(float* __restrict__ deg, int n) {
  int i = blockIdx.x * blockDim.x + threadIdx.x;
  if (i < n) deg[i] = 1.0f;
}

// deg[col[e]] += 1 for each non-self edge
__global__ __launch_bounds__(256) void deg_edge_k(const long long* __restrict__ row,
                                                  const long long* __restrict__ col,
                                                  float* __restrict__ deg, int E) {
  int stride = gridDim.x * blockDim.x;
  for (int e = blockIdx.x * blockDim.x + threadIdx.x; e < E; e += stride) {
    int r = (int)row[e];
    int c = (int)col[e];
    if (r != c) ATOMIC_ADD_F32(&deg[c], 1.0f);
  }
}

// dinv[i] = rsqrt(deg[i])   (deg >= 1 always)
__global__ __launch_bounds__(256) void dinv_k(float* __restrict__ deg, int n) {
  int i = blockIdx.x * blockDim.x + threadIdx.x;
  if (i < n) deg[i] = __frsqrt_rn(deg[i]);
}

// ---------------------------------------------------------------------------
// C[M x NT*16] = A[M x K] @ W[NT*16 x K]^T (+bias).
// One wave32 owns a 16-row strip of A and ALL NT column tiles: each A float2
// load feeds NT v_wmma_f32_16x16x4_f32 ops (independent accumulator chains),
// so A is streamed from HBM exactly once. W (<=96KB) stays L2-resident.
//
// f32 16x16x4 operand layout (ISA 7.12.2):
//  A (16x4):  lanes 0-15 row=lane,  v0=K0,v1=K1 ; lanes 16-31 row=lane-16, v0=K2,v1=K3
//  B (4x16):  lanes 0-15 col=lane,  v0=K0,v1=K1 ; lanes 16-31 col=lane-16, v0=K2,v1=K3
//  C (16x16): v[i] = row i (lanes 0-15) / row i+8 (lanes 16-31), col = lane&15
template <int NT>
__global__ __launch_bounds__(32)
void wmma_gemm_strip(const float* __restrict__ A, const float* __restrict__ W,
                     const float* __restrict__ bias, float* __restrict__ C,
                     int K) {
  const int Nc = NT * 16;
  int mt   = blockIdx.x;
  int lane = threadIdx.x;
  int half = lane >> 4;   // 0 or 1
  int l    = lane & 15;

  const float* ap = A + (size_t)(mt * 16 + l) * K + 2 * half;
  const float* wp = W + (size_t)l * K + 2 * half;   // + nt*16*K per column tile

  v8f acc[NT] = {};
  for (int k = 0; k < K; k += 4) {
    v2f a = *(const v2f*)(ap + k);
#pragma unroll
    for (int nt = 0; nt < NT; ++nt) {
      v2f b = *(const v2f*)(wp + (size_t)nt * 16 * K + k);
      acc[nt] = __builtin_amdgcn_wmma_f32_16x16x4_f32(false, a, false, b,
                                                      (short)0, acc[nt], false, false);
    }
  }

#pragma unroll
  for (int nt = 0; nt < NT; ++nt) {
    float bv = bias ? bias[nt * 16 + l] : 0.0f;
    float* cp = C + (size_t)(mt * 16 + half * 8) * Nc + nt * 16 + l;
#pragma unroll
    for (int i = 0; i < 8; ++i) cp[(size_t)i * Nc] = acc[nt][i] + bv;
  }
}

// ---------------------------------------------------------------------------
// agg[i][f] = dinv[i]^2 * h[i][f] (+ bias[f])   -- self-loop term + optional bias
__global__ void self_init_k(const float* __restrict__ dinv, const float* __restrict__ h,
                            const float* __restrict__ bias, float* __restrict__ agg,
                            int F) {
  int i = blockIdx.x, f = threadIdx.x;
  float d = dinv[i];
  float b = bias ? bias[f] : 0.0f;
  agg[(size_t)i * F + f] = d * d * h[(size_t)i * F + f] + b;
}

// agg[col[e]][f] += dinv[r]*dinv[c] * h[row[e]][f]   (blockDim.x == F: coalesced)
__global__ void edge_prop_k(const long long* __restrict__ row, const long long* __restrict__ col,
                            const float* __restrict__ dinv, const float* __restrict__ h,
                            float* __restrict__ agg, int E, int F) {
  int f = threadIdx.x;
  for (int e = blockIdx.x; e < E; e += gridDim.x) {
    int r = (int)row[e];
    int c = (int)col[e];
    if (r == c) continue;
    float w = dinv[r] * dinv[c];
    ATOMIC_ADD_F32(&agg[(size_t)c * F + f], w * h[(size_t)r * F + f]);
  }
}

// ---------------------------------------------------------------------------
__global__ __launch_bounds__(256) void selu_k(float* __restrict__ h, int n) {
  int i = blockIdx.x * blockDim.x + threadIdx.x;
  if (i >= n) return;
  const float scale = 1.0507009873554805f;
  const float alpha = 1.6732632423543772f;
  float v = h[i];
  h[i] = v > 0.0f ? scale * v : scale * alpha * (expf(v) - 1.0f);
}

// Row-wise log-softmax over 64 cols: one wave32 per row, 2 elems/lane.
__global__ __launch_bounds__(256) void logsoftmax64_k(float* __restrict__ out, int n) {
  int wid  = (blockIdx.x * blockDim.x + threadIdx.x) >> 5;
  int lane = threadIdx.x & 31;
  if (wid >= n) return;
  float* p = out + (size_t)wid * 64;
  float v0 = p[lane], v1 = p[lane + 32];
  float m = fmaxf(v0, v1);
#pragma unroll
  for (int off = 16; off; off >>= 1) m = fmaxf(m, __shfl_xor(m, off));
  float s = expf(v0 - m) + expf(v1 - m);
#pragma unroll
  for (int off = 16; off; off >>= 1) s += __shfl_xor(s, off);
  float ls = logf(s);
  p[lane]      = v0 - m - ls;
  p[lane + 32] = v1 - m - ls;
}

// ---------------------------------------------------------------------------
static inline size_t align256(size_t x) { return (x + 255) & ~(size_t)255; }

extern "C" void kernel_launch(void* const* d_in, const int* in_sizes, int n_in,
                              void* d_out, int out_size, void* d_ws, size_t ws_size,
                              hipStream_t stream) {
  const float*     x    = (const float*)d_in[0];
  const long long* ei   = (const long long*)d_in[1];
  const float*     fc_w = (const float*)d_in[2];
  const float*     fc_b = (const float*)d_in[3];
  const float*     w2   = (const float*)d_in[4];
  const float*     b2   = (const float*)d_in[5];
  float*           out  = (float*)d_out;

  const int HID = in_sizes[3];              // 96
  const int OUT = in_sizes[5];              // 64
  const int IN_DIM = in_sizes[2] / HID;     // 256
  const int N = in_sizes[0] / IN_DIM;       // 50000
  const int E = in_sizes[1] / 2;            // 800000

  const long long* row = ei;
  const long long* col = ei + E;

  char* ws = (char*)d_ws;
  size_t off = 0;
  float* dinv = (float*)(ws + off); off += align256((size_t)N * 4);
  float* h1   = (float*)(ws + off); off += align256((size_t)N * HID * 4);
  float* agg1 = (float*)(ws + off);
  float* h2   = h1;  // h1 is dead after propagate1; reuse for GEMM2 output

  // 1) degree / dinv
  deg_init_k<<<(N + 255) / 256, 256, 0, stream>>>(dinv, N);
  deg_edge_k<<<4096, 256, 0, stream>>>(row, col, dinv, E);
  dinv_k<<<(N + 255) / 256, 256, 0, stream>>>(dinv, N);

  // 2) h1 = x @ fc_w^T + fc_b   (WMMA f32, 16-row strip x 96 cols per wave)
  wmma_gemm_strip<6><<<N / 16, 32, 0, stream>>>(x, fc_w, fc_b, h1, IN_DIM);

  // 3) propagate 1: agg1 = D^-1/2 A D^-1/2 h1
  self_init_k<<<N, HID, 0, stream>>>(dinv, h1, nullptr, agg1, HID);
  edge_prop_k<<<8192, HID, 0, stream>>>(row, col, dinv, h1, agg1, E, HID);

  // 4) SELU in place
  selu_k<<<(N * HID + 255) / 256, 256, 0, stream>>>(agg1, N * HID);

  // 5) h2 = selu(agg1) @ w2^T   (WMMA f32, 16-row strip x 64 cols per wave)
  wmma_gemm_strip<4><<<N / 16, 32, 0, stream>>>(agg1, w2, nullptr, h2, HID);

  // 6) propagate 2 directly into d_out, bias folded into the init
  self_init_k<<<N, OUT, 0, stream>>>(dinv, h2, b2, out, OUT);
  edge_prop_k<<<8192, OUT, 0, stream>>>(row, col, dinv, h2, out, E, OUT);

  // 7) log-softmax rows in place
  logsoftmax64_k<<<(N * 32 + 255) / 256, 256, 0, stream>>>(out, N);
}